// CustomGNN_5592047419419
// MI455X (gfx1250) — compile-verified
//
#include <hip/hip_runtime.h>
#include <math.h>

#define N_NODES 100000
#define N_EDGES 500000
#define HEADS   4
#define HIDC    64
#define N_GRAPHS 512

typedef float v2f __attribute__((ext_vector_type(2)));
typedef float v8f __attribute__((ext_vector_type(8)));

// ---------------- float atomic max (handles negatives, init = -inf) ----------
__device__ __forceinline__ void atomicMaxF(float* addr, float v) {
  if (v >= 0.0f) atomicMax((int*)addr, __float_as_int(v));
  else           atomicMin((unsigned int*)addr, __float_as_uint(v));
}

// ---------------- fill -------------------------------------------------------
__global__ void fill_f32(float* __restrict__ p, float v, int n) {
  int t = blockIdx.x * blockDim.x + threadIdx.x;
  if (t < n) p[t] = v;
}

// ---------------- GEMM: H[N,256] = X[N,64] @ W[256,64]^T  (f32 WMMA) ---------
// One wave computes one 16x16 tile of H. 100000/16 = 6250 M-tiles, 16 N-tiles.
// Grid: 12500 blocks x 256 threads = 100000 waves, exact cover (no divergence,
// EXEC stays all-ones as WMMA requires).
__global__ void gemm_wmma(const float* __restrict__ X,
                          const float* __restrict__ W,
                          float* __restrict__ H) {
  int wave   = blockIdx.x * 8 + (threadIdx.x >> 5);
  int lane   = threadIdx.x & 31;
  int tile_m = wave >> 4;          // 0..6249
  int tile_n = wave & 15;          // 0..15
  int m      = tile_m * 16 + (lane & 15);   // A row for this lane
  int ncol   = tile_n * 16 + (lane & 15);   // B col for this lane
  int kh     = (lane >> 4) * 2;             // K sub-offset per lane half

  const float* __restrict__ xrow = X + (size_t)m * 64;
  const float* __restrict__ wrow = W + (size_t)ncol * 64;

  v8f c = {};
#pragma unroll
  for (int k = 0; k < 64; k += 4) {
    v2f a, b;
    a.x = xrow[k + kh];
    a.y = xrow[k + kh + 1];
    b.x = wrow[k + kh];
    b.y = wrow[k + kh + 1];
    c = __builtin_amdgcn_wmma_f32_16x16x4_f32(false, a, false, b,
                                              (short)0, c, false, false);
  }

  int mbase = tile_m * 16 + (lane >> 4) * 8;
  int nc    = tile_n * 16 + (lane & 15);
#pragma unroll
  for (int r = 0; r < 8; ++r)
    H[(size_t)(mbase + r) * 256 + nc] = c[r];
}

// ---------------- per-node attention coefficients ----------------------------
// a_s[n,h] = sum_c H[n,h,c]*att_src[h,c] ; a_d likewise.
__global__ void attn_coef(const float* __restrict__ H,
                          const float* __restrict__ asrc,
                          const float* __restrict__ adst,
                          float* __restrict__ a_s,
                          float* __restrict__ a_d) {
  int t = blockIdx.x * blockDim.x + threadIdx.x;
  if (t >= N_NODES * HEADS) return;
  int n = t >> 2, h = t & 3;
  const float* hp = H + (size_t)n * 256 + h * 64;
  const float* sp = asrc + h * 64;
  const float* dp = adst + h * 64;
  float ss = 0.f, dd = 0.f;
#pragma unroll 8
  for (int c = 0; c < 64; ++c) { float v = hp[c]; ss += v * sp[c]; dd += v * dp[c]; }
  a_s[t] = ss;
  a_d[t] = dd;
}

// ---------------- edge pass 1: segment max of leaky-relu logits --------------
__global__ void edge_max(const int* __restrict__ ei,
                         const float* __restrict__ a_s,
                         const float* __restrict__ a_d,
                         float* __restrict__ emax) {
  int t = blockIdx.x * blockDim.x + threadIdx.x;
  if (t >= N_EDGES * HEADS) return;
  int e = t >> 2, h = t & 3;
  int row = ei[e], col = ei[N_EDGES + e];
  float v = a_s[row * 4 + h] + a_d[col * 4 + h];
  v = v >= 0.f ? v : 0.2f * v;
  atomicMaxF(&emax[col * 4 + h], v);
}

// ---------------- edge pass 2: segment sum of exp(e - max) -------------------
__global__ void edge_sum(const int* __restrict__ ei,
                         const float* __restrict__ a_s,
                         const float* __restrict__ a_d,
                         const float* __restrict__ emax,
                         float* __restrict__ psum) {
  int t = blockIdx.x * blockDim.x + threadIdx.x;
  if (t >= N_EDGES * HEADS) return;
  int e = t >> 2, h = t & 3;
  int row = ei[e], col = ei[N_EDGES + e];
  float v = a_s[row * 4 + h] + a_d[col * 4 + h];
  v = v >= 0.f ? v : 0.2f * v;
  float p = __expf(v - emax[col * 4 + h]);
  atomicAdd(&psum[col * 4 + h], p);
}

// ---------------- edge pass 3: weighted message scatter-add ------------------
// out[col, c] += 0.25 * sum_h alpha[e,h] * H[row, h, c]   (head-mean fused)
__global__ void edge_msg(const int* __restrict__ ei,
                         const float* __restrict__ a_s,
                         const float* __restrict__ a_d,
                         const float* __restrict__ emax,
                         const float* __restrict__ psum,
                         const float* __restrict__ H,
                         float* __restrict__ out) {
  int t = blockIdx.x * blockDim.x + threadIdx.x;
  if (t >= N_EDGES * 64) return;
  int e = t >> 6, c = t & 63;
  int row = ei[e], col = ei[N_EDGES + e];
  float acc = 0.f;
#pragma unroll
  for (int h = 0; h < 4; ++h) {
    float v = a_s[row * 4 + h] + a_d[col * 4 + h];
    v = v >= 0.f ? v : 0.2f * v;
    float p = __expf(v - emax[col * 4 + h]);
    float alpha = p / (psum[col * 4 + h] + 1e-8f);
    acc += H[(size_t)row * 256 + h * 64 + c] * alpha;
  }
  atomicAdd(&out[(size_t)col * 64 + c], 0.25f * acc);
}

// ---------------- BN(eval) + ReLU, in place over [N,64] ----------------------
__global__ void bn_relu(float* __restrict__ buf,
                        const float* __restrict__ gamma,
                        const float* __restrict__ beta,
                        const float* __restrict__ mean,
                        const float* __restrict__ var) {
  int t = blockIdx.x * blockDim.x + threadIdx.x;
  if (t >= N_NODES * 64) return;
  int c = t & 63;
  float y = (buf[t] - mean[c]) * rsqrtf(var[c] + 1e-5f) * gamma[c] + beta[c];
  buf[t] = y > 0.f ? y : 0.f;
}

// ---------------- global mean pool: per-graph sums + counts ------------------
__global__ void pool_kernel(const float* __restrict__ in,
                            const int* __restrict__ batch,
                            float* __restrict__ gsum,
                            float* __restrict__ gcnt) {
  int t = blockIdx.x * blockDim.x + threadIdx.x;
  if (t >= N_NODES * 64) return;
  int n = t >> 6, c = t & 63;
  int g = batch[n];
  atomicAdd(&gsum[g * 64 + c], in[t]);
  if (c == 0) atomicAdd(&gcnt[g], 1.0f);
}

// ---------------- final FC: out[g,o] = (gsum[g]/cnt) . fc_w[o] + fc_b[o] -----
__global__ void final_fc(const float* __restrict__ gsum,
                         const float* __restrict__ gcnt,
                         const float* __restrict__ fcw,
                         const float* __restrict__ fcb,
                         float* __restrict__ out) {
  int t = blockIdx.x * blockDim.x + threadIdx.x;
  if (t >= N_GRAPHS * 2) return;
  int g = t >> 1, o = t & 1;
  float inv = 1.0f / fmaxf(gcnt[g], 1.0f);
  float acc = 0.f;
#pragma unroll 8
  for (int c = 0; c < 64; ++c)
    acc += gsum[g * 64 + c] * fcw[o * 64 + c];
  out[t] = acc * inv + fcb[o];
}

// =============================================================================
extern "C" void kernel_launch(void* const* d_in, const int* in_sizes, int n_in,
                              void* d_out, int out_size, void* d_ws, size_t ws_size,
                              hipStream_t stream) {
  const float* x        = (const float*)d_in[0];
  const int*   ei       = (const int*)  d_in[1];   // [2, E]
  const int*   batch    = (const int*)  d_in[2];   // [N]
  const float* Ws       = (const float*)d_in[3];   // [3, 256, 64]
  const float* att_src  = (const float*)d_in[4];   // [3, 4, 64]
  const float* att_dst  = (const float*)d_in[5];   // [3, 4, 64]
  const float* bn_gamma = (const float*)d_in[6];   // [2, 64]
  const float* bn_beta  = (const float*)d_in[7];
  const float* bn_mean  = (const float*)d_in[8];
  const float* bn_var   = (const float*)d_in[9];
  const float* fc_w     = (const float*)d_in[10];  // [2, 64]
  const float* fc_b     = (const float*)d_in[11];  // [2]
  float* out = (float*)d_out;                      // [512, 2]

  // ---- workspace carve-up (floats) ----
  float* ws    = (float*)d_ws;
  float* H     = ws;                                   // N*256
  float* nodeA = H     + (size_t)N_NODES * 256;        // N*64
  float* nodeB = nodeA + (size_t)N_NODES * 64;         // N*64
  float* a_s   = nodeB + (size_t)N_NODES * 64;         // N*4
  float* a_d   = a_s   + (size_t)N_NODES * 4;          // N*4
  float* emax  = a_d   + (size_t)N_NODES * 4;          // N*4
  float* psum  = emax  + (size_t)N_NODES * 4;          // N*4
  float* gsum  = psum  + (size_t)N_NODES * 4;          // 512*64
  float* gcnt  = gsum  + (size_t)N_GRAPHS * 64;        // 512

  const int TB = 256;
  const int nh_blocks  = (N_NODES * HEADS + TB - 1) / TB;
  const int nc_blocks  = (N_NODES * 64 + TB - 1) / TB;
  const int eh_blocks  = (N_EDGES * HEADS + TB - 1) / TB;
  const int ec_blocks  = (N_EDGES * 64 + TB - 1) / TB;

  const float* cur = x;
  float* bufs[3] = { nodeA, nodeB, nodeA };

  for (int layer = 0; layer < 3; ++layer) {
    float* outbuf = bufs[layer];

    // dense projection via f32 WMMA
    gemm_wmma<<<12500, 256, 0, stream>>>(cur, Ws + (size_t)layer * 256 * 64, H);

    // per-node attention scalars
    attn_coef<<<nh_blocks, TB, 0, stream>>>(H,
        att_src + layer * HEADS * 64, att_dst + layer * HEADS * 64, a_s, a_d);

    // init softmax state + output accumulator
    fill_f32<<<nh_blocks, TB, 0, stream>>>(emax, -INFINITY, N_NODES * HEADS);
    fill_f32<<<nh_blocks, TB, 0, stream>>>(psum, 0.0f, N_NODES * HEADS);
    fill_f32<<<nc_blocks, TB, 0, stream>>>(outbuf, 0.0f, N_NODES * 64);

    // segment softmax + message passing
    edge_max<<<eh_blocks, TB, 0, stream>>>(ei, a_s, a_d, emax);
    edge_sum<<<eh_blocks, TB, 0, stream>>>(ei, a_s, a_d, emax, psum);
    edge_msg<<<ec_blocks, TB, 0, stream>>>(ei, a_s, a_d, emax, psum, H, outbuf);

    if (layer < 2)
      bn_relu<<<nc_blocks, TB, 0, stream>>>(outbuf,
          bn_gamma + layer * 64, bn_beta + layer * 64,
          bn_mean + layer * 64,  bn_var + layer * 64);

    cur = outbuf;
  }

  // global mean pool + FC
  fill_f32<<<(N_GRAPHS * 64 + TB - 1) / TB, TB, 0, stream>>>(gsum, 0.0f, N_GRAPHS * 64);
  fill_f32<<<(N_GRAPHS + TB - 1) / TB, TB, 0, stream>>>(gcnt, 0.0f, N_GRAPHS);
  pool_kernel<<<nc_blocks, TB, 0, stream>>>(cur, batch, gsum, gcnt);
  final_fc<<<(N_GRAPHS * 2 + TB - 1) / TB, TB, 0, stream>>>(gsum, gcnt, fc_w, fc_b, out);
}